// GraphProjection_78709570667187
// MI455X (gfx1250) — compile-verified
//
#include <hip/hip_runtime.h>
#include <stdint.h>
#include <stddef.h>

// ---------------- problem constants ----------------
#define NPTS   50000
#define BATCH  4

#define C1 64
#define S1 56
#define SS1 (S1*S1)          // 3136
#define C2 128
#define S2 28
#define SS2 (S2*S2)          // 784
#define C3 256
#define S3 14
#define SS3 (S3*S3)          // 196
#define C4 512
#define S4 7
#define SS4 (S4*S4)          // 49

#define ROW 963              // 3 + 64 + 128 + 256 + 512

// LDS layout (dwords). f3 padded to odd stride 197 -> conflict-free banks.
#define F3_STRIDE 197
#define F3_DW (C3*F3_STRIDE) // 50432
#define F4_DW (C4*SS4)       // 25088
#define LDS_DW (F3_DW + F4_DW)           // 75520 dwords
#define LDS_BYTES ((size_t)LDS_DW * 4)   // 302080 B <= 320KB WGP LDS

#define THREADS 1024
#define PTS_PER_BLOCK 1024

// ---------------- gfx1250 async-to-LDS helpers ----------------
#ifndef __has_builtin
#define __has_builtin(x) 0
#endif

#if __has_builtin(__builtin_amdgcn_global_load_async_to_lds_b32)
#define HAVE_ASYNC32 1
#endif
#if __has_builtin(__builtin_amdgcn_global_load_async_to_lds_b128)
#define HAVE_ASYNC128 1
#endif

typedef int v4i __attribute__((vector_size(16)));

// Pointer casts to the builtins' expected types:
//  arg0: address_space(1) ("__device__") int* / v4i*  -> global source
//  arg1: address_space(3) ("__shared__") int* / v4i*  -> LDS destination
#define AS1_I32(p)  ((__attribute__((address_space(1))) int*)(p))
#define AS3_I32(p)  ((__attribute__((address_space(3))) int*)(p))
#define AS1_V4I(p)  ((__attribute__((address_space(1))) v4i*)(p))
#define AS3_V4I(p)  ((__attribute__((address_space(3))) v4i*)(p))

__device__ __forceinline__ void async_copy_b32(float* l, const float* g) {
#if defined(HAVE_ASYNC32)
    __builtin_amdgcn_global_load_async_to_lds_b32(AS1_I32(g), AS3_I32(l), 0, 0);
#else
    *l = *g;
#endif
}

__device__ __forceinline__ void async_copy_b128(float4* l, const float4* g) {
#if defined(HAVE_ASYNC128)
    __builtin_amdgcn_global_load_async_to_lds_b128(AS1_V4I(g), AS3_V4I(l), 0, 0);
#else
    *l = *g;
#endif
}

__device__ __forceinline__ void wait_async_all() {
#if defined(HAVE_ASYNC32) || defined(HAVE_ASYNC128)
#if __has_builtin(__builtin_amdgcn_s_wait_asynccnt)
    __builtin_amdgcn_s_wait_asynccnt(0);
#else
    asm volatile("s_wait_asynccnt 0" ::: "memory");
#endif
#endif
}

// ---------------- bilinear weight/offset setup ----------------
struct Corner { int o11, o21, o12, o22; float wa, wb, wc, wd; };

__device__ __forceinline__ Corner bilinear(float h, float w, float scale, int s) {
    float x = fminf(fmaxf(h * scale, 0.0f), (float)(s - 1));
    float y = fminf(fmaxf(w * scale, 0.0f), (float)(s - 1));
    float x1f = floorf(x), x2f = ceilf(x);
    float y1f = floorf(y), y2f = ceilf(y);
    float dx1 = x - x1f, dx2 = x2f - x;
    float dy1 = y - y1f, dy2 = y2f - y;
    Corner c;
    c.wa = dx2 * dy2;   // Q11
    c.wb = dx1 * dy2;   // Q21
    c.wc = dx2 * dy1;   // Q12
    c.wd = dx1 * dy1;   // Q22
    int x1 = (int)x1f, x2 = (int)x2f, y1 = (int)y1f, y2 = (int)y2f;
    c.o11 = x1 * s + y1;
    c.o21 = x2 * s + y1;
    c.o12 = x1 * s + y2;
    c.o22 = x2 * s + y2;
    return c;
}

// ---------------- transpose [B,C,SS] -> [B,SS,C] (channel-last) ----------------
__global__ void transpose_feat_kernel(const float* __restrict__ src,
                                      float* __restrict__ dst,
                                      unsigned C, unsigned SS) {
    unsigned total = BATCH * C * SS;
    for (unsigned i = blockIdx.x * blockDim.x + threadIdx.x; i < total;
         i += gridDim.x * blockDim.x) {
        unsigned c   = i % C;
        unsigned rem = i / C;
        unsigned p   = rem % SS;
        unsigned b   = rem / SS;
        dst[i] = src[((size_t)b * C + c) * SS + p];
    }
}

// ---------------- main projection kernel ----------------
__global__ __launch_bounds__(THREADS)
void graph_projection_kernel(const float* __restrict__ coord,   // batch 0 coords, [NPTS,3]
                             const float* __restrict__ f1,      // [B,64,56,56]
                             const float* __restrict__ f2,      // [B,128,28,28]
                             const float* __restrict__ f3,      // [B,256,14,14]
                             const float* __restrict__ f4,      // [B,512,7,7]
                             const float* __restrict__ f1t,     // [B,3136,64] (optional)
                             const float* __restrict__ f2t,     // [B,784,128] (optional)
                             int use_transposed,
                             float* __restrict__ out)           // [B,NPTS,963]
{
    extern __shared__ float lds[];
    float* l3 = lds;            // C3 channels, stride F3_STRIDE dwords (odd -> no conflicts)
    float* l4 = lds + F3_DW;    // C4 channels, stride 49 dwords (odd -> no conflicts)

    const int b   = blockIdx.y;
    const int tid = threadIdx.x;

    // ---- stage f3 (padded) + f4 (flat b128) into LDS via async copies ----
    {
        const float* G3 = f3 + (size_t)b * (C3 * SS3);
        for (int i = tid; i < C3 * SS3; i += THREADS) {
            int c = i / SS3;
            int r = i - c * SS3;
            async_copy_b32(&l3[c * F3_STRIDE + r], &G3[i]);
        }
        const float4* G4 = (const float4*)(f4 + (size_t)b * (C4 * SS4));
        float4* L4 = (float4*)l4;
        for (int i = tid; i < (C4 * SS4) / 4; i += THREADS) {
            async_copy_b128(&L4[i], &G4[i]);
        }
    }
    wait_async_all();
    __syncthreads();

    const int lane   = tid & 31;
    const int wave   = tid >> 5;
    const int nwaves = THREADS >> 5;

    for (int p = wave; p < PTS_PER_BLOCK; p += nwaves) {
        const int n = blockIdx.x * PTS_PER_BLOCK + p;
        if (n >= NPTS) break;

        const float X = coord[3 * n + 0];
        const float Y = coord[3 * n + 1];
        const float Z = coord[3 * n + 2];
        const float h = fminf(fmaxf((250.0f * -Y) / (-Z) + 112.0f, 0.0f), 223.0f);
        const float w = fminf(fmaxf((250.0f *  X) / (-Z) + 112.0f, 0.0f), 223.0f);

        float* __restrict__ orow = out + ((size_t)b * NPTS + n) * ROW;

        if (lane < 3) {
            float cv = (lane == 0) ? X : (lane == 1) ? Y : Z;
            __builtin_nontemporal_store(cv, &orow[lane]);
        }

        // ---- level 1: 64ch @ 56x56 (scale 1/4), L2-served ----
        {
            Corner k = bilinear(h, w, 0.25f, S1);
            if (use_transposed) {
                const float* base = f1t + (size_t)b * (SS1 * C1);
                const float* q11 = base + (size_t)k.o11 * C1;
                const float* q21 = base + (size_t)k.o21 * C1;
                const float* q12 = base + (size_t)k.o12 * C1;
                const float* q22 = base + (size_t)k.o22 * C1;
                #pragma unroll
                for (int c = lane; c < C1; c += 32) {
                    float v = k.wa * q11[c] + k.wb * q21[c] + k.wc * q12[c] + k.wd * q22[c];
                    __builtin_nontemporal_store(v, &orow[3 + c]);
                }
            } else {
                const float* base = f1 + (size_t)b * (C1 * SS1);
                #pragma unroll
                for (int c = lane; c < C1; c += 32) {
                    const float* q = base + (size_t)c * SS1;
                    float v = k.wa * q[k.o11] + k.wb * q[k.o21] + k.wc * q[k.o12] + k.wd * q[k.o22];
                    __builtin_nontemporal_store(v, &orow[3 + c]);
                }
            }
        }

        // ---- level 2: 128ch @ 28x28 (scale 1/8), L2-served ----
        {
            Corner k = bilinear(h, w, 0.125f, S2);
            if (use_transposed) {
                const float* base = f2t + (size_t)b * (SS2 * C2);
                const float* q11 = base + (size_t)k.o11 * C2;
                const float* q21 = base + (size_t)k.o21 * C2;
                const float* q12 = base + (size_t)k.o12 * C2;
                const float* q22 = base + (size_t)k.o22 * C2;
                #pragma unroll
                for (int c = lane; c < C2; c += 32) {
                    float v = k.wa * q11[c] + k.wb * q21[c] + k.wc * q12[c] + k.wd * q22[c];
                    __builtin_nontemporal_store(v, &orow[3 + C1 + c]);
                }
            } else {
                const float* base = f2 + (size_t)b * (C2 * SS2);
                #pragma unroll
                for (int c = lane; c < C2; c += 32) {
                    const float* q = base + (size_t)c * SS2;
                    float v = k.wa * q[k.o11] + k.wb * q[k.o21] + k.wc * q[k.o12] + k.wd * q[k.o22];
                    __builtin_nontemporal_store(v, &orow[3 + C1 + c]);
                }
            }
        }

        // ---- level 3: 256ch @ 14x14 (scale 1/16), LDS-served, stride 197 ----
        {
            Corner k = bilinear(h, w, 0.0625f, S3);
            #pragma unroll
            for (int c = lane; c < C3; c += 32) {
                const float* q = &l3[c * F3_STRIDE];
                float v = k.wa * q[k.o11] + k.wb * q[k.o21] + k.wc * q[k.o12] + k.wd * q[k.o22];
                __builtin_nontemporal_store(v, &orow[3 + C1 + C2 + c]);
            }
        }

        // ---- level 4: 512ch @ 7x7 (scale 1/32), LDS-served, stride 49 ----
        {
            Corner k = bilinear(h, w, 0.03125f, S4);
            #pragma unroll
            for (int c = lane; c < C4; c += 32) {
                const float* q = &l4[c * SS4];
                float v = k.wa * q[k.o11] + k.wb * q[k.o21] + k.wc * q[k.o12] + k.wd * q[k.o22];
                __builtin_nontemporal_store(v, &orow[3 + C1 + C2 + C3 + c]);
            }
        }
    }
}

// ---------------- host launcher ----------------
extern "C" void kernel_launch(void* const* d_in, const int* in_sizes, int n_in,
                              void* d_out, int out_size, void* d_ws, size_t ws_size,
                              hipStream_t stream) {
    const float* coord = (const float*)d_in[0]; // [4,50000,3]; only batch 0 used
    const float* f1    = (const float*)d_in[1];
    const float* f2    = (const float*)d_in[2];
    const float* f3    = (const float*)d_in[3];
    const float* f4    = (const float*)d_in[4];
    float* out = (float*)d_out;

    // Channel-last copies of f1/f2 in scratch, if it fits.
    const size_t f1t_elems = (size_t)BATCH * SS1 * C1;   // 802,816
    const size_t f2t_elems = (size_t)BATCH * SS2 * C2;   // 401,408
    const size_t need = (f1t_elems + f2t_elems) * sizeof(float); // ~4.6 MB
    int use_t = (d_ws != nullptr && ws_size >= need) ? 1 : 0;
    float* f1t = (float*)d_ws;
    float* f2t = f1t + f1t_elems;

    if (use_t) {
        transpose_feat_kernel<<<2048, 256, 0, stream>>>(f1, f1t, C1, SS1);
        transpose_feat_kernel<<<2048, 256, 0, stream>>>(f2, f2t, C2, SS2);
    }

    (void)hipFuncSetAttribute((const void*)graph_projection_kernel,
                              hipFuncAttributeMaxDynamicSharedMemorySize,
                              (int)LDS_BYTES);

    dim3 grid((NPTS + PTS_PER_BLOCK - 1) / PTS_PER_BLOCK, BATCH);
    graph_projection_kernel<<<grid, THREADS, LDS_BYTES, stream>>>(
        coord, f1, f2, f3, f4, f1t, f2t, use_t, out);

    (void)in_sizes; (void)n_in; (void)out_size;
}